// Net_30382598652211
// MI455X (gfx1250) — compile-verified
//
#include <hip/hip_runtime.h>
#include <hip/hip_bf16.h>

typedef __attribute__((ext_vector_type(2))) float v2f;
typedef __attribute__((ext_vector_type(8))) float v8f;

// Problem constants
#define B_SZ   4096
#define IN_F   4096
#define CAP    512
#define COMP   256
#define OUT_F  128
#define N_CAPS 16
#define CONV_OC 8
#define CONV_K  5
#define QKV_E  768          // 3*COMP

// CDNA5 async global->LDS path (ASYNCcnt-tracked), guarded so the host pass
// and older toolchains fall back to register staging.
// Probe: param 0 is an int4 ('__vector_size__(4*sizeof(int)) int') pointer in
// AS(1); LDS side is the AS(3) counterpart.
#if __has_builtin(__builtin_amdgcn_global_load_async_to_lds_b128) && \
    __has_builtin(__builtin_amdgcn_s_wait_asynccnt)
#define USE_ASYNC_LDS 1
typedef int v4i __attribute__((vector_size(4 * sizeof(int))));
typedef v4i __attribute__((address_space(1))) gv4i;
typedef v4i __attribute__((address_space(3))) sv4i;
#else
#define USE_ASYNC_LDS 0
#endif

// D = A(16x4) * B(4x16) + C, fp32 WMMA (native CDNA5 fp32 matrix path)
__device__ __forceinline__ v8f wmma4_f32(v2f a, v2f b, v8f c) {
    return __builtin_amdgcn_wmma_f32_16x16x4_f32(
        /*neg_a=*/false, a, /*neg_b=*/false, b,
        /*c_mod=*/(short)0, c, /*reuse_a=*/false, /*reuse_b=*/false);
}

// ---------------------------------------------------------------------------
// Generic fp32 WMMA GEMM:  C[z] = act( A * B[z] + bias[z] )
//   A: M x K row-major (lda), shared across z
//   B: K x N row-major (ldb), base + z*B_zstride
//   C: row-major (ldc), base + z*C_zstride
// Tile: 128(M) x 128(N) per 256-thread block (8 waves, 16x128 strip/wave),
// K-chunk 32.
//   A LDS: row-major, stride 36  -> a.x/a.y contiguous => ds_load_2addr_b64,
//          halves (cols k / k+2) provably bank-disjoint.
//   B LDS: K-PAIR INTERLEAVED  Bs[p][n][0..1] = B[2p][n], B[2p+1][n]
//          -> each fragment is one 8B load (no v_mov packing).  Pair-row
//          stride 288 == 32 (mod 64) banks => the two 16-lane halves (pair
//          rows ks/2 and ks/2+1) hit disjoint bank windows.
// A tile staged with async global->LDS (no VGPR round-trip); B staged via
// registers (needs the pair interleave shuffle).  B fragments of each K-step
// are loaded into 8 distinct registers before the 8 WMMAs so the backend can
// overlap ds_load latency with the WMMA chain (no same-register WAR stalls).
// ---------------------------------------------------------------------------
#define TM 128
#define TN 128
#define TK 32
#define ASTRIDE 36
#define BPSTRIDE 288

__global__ __launch_bounds__(256) void wmma_gemm_f32(
    const float* __restrict__ A, int lda,
    const float* __restrict__ B, int ldb, long long B_zstride,
    float* __restrict__ C, int ldc, long long C_zstride,
    const float* __restrict__ bias, int bias_zstride,
    int K, int apply_relu)
{
    __shared__ __align__(16) float As[TM * ASTRIDE];          // 128 x 32 rows
    __shared__ __align__(16) float Bs[(TK / 2) * BPSTRIDE];   // 16 pair-rows x 256

    const int tid  = threadIdx.x;
    const int wave = tid >> 5;          // wave32
    const int lane = tid & 31;
    const int lm   = lane & 15;         // M index (A) / N index (B) in tile
    const int lk2  = (lane >> 4) * 2;   // K pair base: 0 or 2
    const int half = lane >> 4;
    const int z    = blockIdx.z;

    const int m0 = blockIdx.x * TM;
    const int n0 = blockIdx.y * TN;
    const int wm = wave * 16;           // wave's row strip inside tile

    const float* Bz = B + (long long)z * B_zstride;

    // Hoisted per-thread staging pointers (advance by TK / TK*ldb per chunk).
    // A units: 128x32 tile = 1024 float4, 8 per row; 4 units/thread.
    const float* pA[4]; int lofsA[4];
    #pragma unroll
    for (int i = 0; i < 4; ++i) {
        int e  = tid + i * 256;
        int r  = e >> 3;
        int c4 = e & 7;
        pA[i]    = A + (long long)(m0 + r) * lda + c4 * 4;
        lofsA[i] = r * ASTRIDE + c4 * 4;
    }
    // B units: 16 pair-rows x 32 col-quads = 512 units; 2 units/thread.
    const float* pB[2]; int lofsB[2];
    #pragma unroll
    for (int i = 0; i < 2; ++i) {
        int u  = tid + i * 256;
        int p  = u >> 5;
        int n4 = u & 31;
        pB[i]    = Bz + (long long)(2 * p) * ldb + n0 + n4 * 4;
        lofsB[i] = p * BPSTRIDE + n4 * 8;
    }

    v8f acc[8] = {};

    for (int k0 = 0; k0 < K; k0 += TK) {
        __syncthreads();
        // ---- Stage A tile (async direct-to-LDS when available) ----
        #pragma unroll
        for (int i = 0; i < 4; ++i) {
#if USE_ASYNC_LDS
            __builtin_amdgcn_global_load_async_to_lds_b128(
                (gv4i*)pA[i], (sv4i*)(&As[lofsA[i]]), 0, 0);
#else
            float4 av = *(const float4*)pA[i];
            *(float4*)(&As[lofsA[i]]) = av;
#endif
            pA[i] += TK;
        }
        // ---- Stage B tile, K-pair interleaved ----
        #pragma unroll
        for (int i = 0; i < 2; ++i) {
            float4 b0 = *(const float4*)(pB[i]);
            float4 b1 = *(const float4*)(pB[i] + ldb);
            *(float4*)(&Bs[lofsB[i]])     = make_float4(b0.x, b1.x, b0.y, b1.y);
            *(float4*)(&Bs[lofsB[i] + 4]) = make_float4(b0.z, b1.z, b0.w, b1.w);
            pB[i] += (long long)TK * ldb;
        }
        // ---- Prefetch next chunk's lines (global_prefetch_b8) ----
        if (k0 + TK < K) {
            #pragma unroll
            for (int i = 0; i < 4; ++i) __builtin_prefetch(pA[i], 0, 1);
            #pragma unroll
            for (int i = 0; i < 2; ++i) __builtin_prefetch(pB[i], 0, 1);
        }
#if USE_ASYNC_LDS
        __builtin_amdgcn_s_wait_asynccnt(0);
#endif
        __syncthreads();

        #pragma unroll
        for (int ks = 0; ks < TK; ks += 4) {
            v2f a = *(const v2f*)(&As[(wm + lm) * ASTRIDE + ks + lk2]);
            const int prow = (ks >> 1) + half;
            // Load all 8 B fragments into distinct regs first so ds_load
            // latency overlaps the WMMA chain instead of stalling per pair.
            v2f bfr[8];
            #pragma unroll
            for (int t = 0; t < 8; ++t)
                bfr[t] = *(const v2f*)(&Bs[prow * BPSTRIDE + 32 * t + 2 * lm]);
            #pragma unroll
            for (int t = 0; t < 8; ++t)
                acc[t] = wmma4_f32(a, bfr[t], acc[t]);
        }
    }

    // Epilogue: C/D layout — lane holds N = lm, rows M = j + 8*(lane>>4)
    float* Cz = C + (long long)z * C_zstride;
    const int mrow = m0 + wm + 8 * half;
    #pragma unroll
    for (int t = 0; t < 8; ++t) {
        int n = n0 + t * 16 + lm;
        float bv = bias ? bias[(long long)z * bias_zstride + n] : 0.0f;
        #pragma unroll
        for (int j = 0; j < 8; ++j) {
            float v = acc[t][j] + bv;
            if (apply_relu) v = fmaxf(v, 0.0f);
            Cz[(long long)(mrow + j) * ldc + n] = v;
        }
    }
}

// ---------------------------------------------------------------------------
// In-place row L2 normalization: rows of `ncols` floats, one block per row.
// ---------------------------------------------------------------------------
__global__ __launch_bounds__(128) void l2norm_rows(float* __restrict__ data, int ncols)
{
    float* p = data + (long long)blockIdx.x * ncols;
    float s = 0.0f;
    for (int c = threadIdx.x; c < ncols; c += 128) { float v = p[c]; s += v * v; }
    #pragma unroll
    for (int off = 16; off > 0; off >>= 1) s += __shfl_xor(s, off, 32);
    __shared__ float wsum[4];
    if ((threadIdx.x & 31) == 0) wsum[threadIdx.x >> 5] = s;
    __syncthreads();
    float tot = wsum[0] + wsum[1] + wsum[2] + wsum[3];
    float scale = 1.0f / fmaxf(sqrtf(tot), 1e-12f);
    for (int c = threadIdx.x; c < ncols; c += 128) p[c] *= scale;
}

// ---------------------------------------------------------------------------
// Attention: one wave per batch element (4 waves / block).
//   scores = q(16x256) @ k^T(256x16)  -> 64 fp32 WMMAs
//   softmax over 16 cols (16 lanes, via LDS)
//   cf_raw = attn(16x16) @ v(16x256)  -> 16 tiles x 4 WMMAs
//   L2-normalize rows, store cf to d_out.
// ---------------------------------------------------------------------------
#define SC_STRIDE 17
#define CF_STRIDE 260
#define WAVE_LDS  (16 * SC_STRIDE + 16 * CF_STRIDE)   // 4432 floats

__global__ __launch_bounds__(128) void attn_kernel(
    const float* __restrict__ qkv,      // (B*16, 768): [q|k|v]
    float* __restrict__ cf_out)         // (B, 16, 256)
{
    __shared__ float sm[4 * WAVE_LDS];
    const int wave = threadIdx.x >> 5;
    const int lane = threadIdx.x & 31;
    const int lm   = lane & 15;
    const int lk2  = (lane >> 4) * 2;
    const int b    = blockIdx.x * 4 + wave;

    float* sc = sm + wave * WAVE_LDS;        // 16x17 scores / attn
    float* cf = sc + 16 * SC_STRIDE;         // 16x260 raw context

    const float* qb = qkv + (long long)b * 16 * QKV_E;

    // scores = q @ k^T : A[m][kk]=q[lm][ks+lk2], B[kk][n]=k[lm][ks+lk2]
    v8f acc = {};
    #pragma unroll 8
    for (int ks = 0; ks < COMP; ks += 4) {
        v2f a, bb;
        a.x  = qb[lm * QKV_E + ks + lk2];
        a.y  = qb[lm * QKV_E + ks + lk2 + 1];
        bb.x = qb[lm * QKV_E + COMP + ks + lk2];
        bb.y = qb[lm * QKV_E + COMP + ks + lk2 + 1];
        acc = wmma4_f32(a, bb, acc);
    }
    #pragma unroll
    for (int j = 0; j < 8; ++j)
        sc[(j + 8 * (lane >> 4)) * SC_STRIDE + lm] = acc[j];
    __syncthreads();

    // softmax over columns, one lane per query row
    if (lane < 16) {
        float mx = -3.4e38f;
        for (int n = 0; n < 16; ++n) mx = fmaxf(mx, sc[lane * SC_STRIDE + n]);
        float ssum = 0.0f;
        for (int n = 0; n < 16; ++n) {
            float e = __expf(sc[lane * SC_STRIDE + n] - mx);
            sc[lane * SC_STRIDE + n] = e;
            ssum += e;
        }
        float inv = 1.0f / ssum;
        for (int n = 0; n < 16; ++n) sc[lane * SC_STRIDE + n] *= inv;
    }
    __syncthreads();

    // cf_raw = attn @ v  (16 N-tiles of 16, K=16 -> 4 WMMA k-steps each)
    #pragma unroll
    for (int t = 0; t < 16; ++t) {
        v8f a2 = {};
        #pragma unroll
        for (int ks = 0; ks < 16; ks += 4) {
            v2f a, bb;
            a.x  = sc[lm * SC_STRIDE + ks + lk2];
            a.y  = sc[lm * SC_STRIDE + ks + lk2 + 1];
            bb.x = qb[(long long)(ks + lk2)     * QKV_E + 2 * COMP + t * 16 + lm];
            bb.y = qb[(long long)(ks + lk2 + 1) * QKV_E + 2 * COMP + t * 16 + lm];
            a2 = wmma4_f32(a, bb, a2);
        }
        #pragma unroll
        for (int j = 0; j < 8; ++j)
            cf[(j + 8 * (lane >> 4)) * CF_STRIDE + t * 16 + lm] = a2[j];
    }
    __syncthreads();

    // fused L2 norm + store
    if (lane < 16) {
        float ssum = 0.0f;
        for (int d = 0; d < COMP; ++d) { float v = cf[lane * CF_STRIDE + d]; ssum += v * v; }
        float scale = 1.0f / fmaxf(sqrtf(ssum), 1e-12f);
        float* out = cf_out + ((long long)b * 16 + lane) * COMP;
        for (int d = 0; d < COMP; ++d) out[d] = cf[lane * CF_STRIDE + d] * scale;
    }
}

// ---------------------------------------------------------------------------
// Conv1d (valid, K=5, OC=8 over 256 channels x 16 positions) + dense head.
// One 128-thread block per batch element; tiny FLOP share -> plain VALU.
// ---------------------------------------------------------------------------
__global__ __launch_bounds__(128) void conv_head(
    const float* __restrict__ cf,       // (B, 16, 256) normalized
    const float* __restrict__ conv_w,   // (8, 256, 5)
    const float* __restrict__ conv_b,   // (8,)
    const float* __restrict__ c_w,      // (128, 96)
    const float* __restrict__ c_b,      // (128,)
    float* __restrict__ res)            // (B, 128)
{
    __shared__ float cfl[16 * CF_STRIDE];
    __shared__ float convl[CONV_OC * 12];
    const int b   = blockIdx.x;
    const int tid = threadIdx.x;

    const float* src = cf + (long long)b * 16 * COMP;
    for (int i = tid; i < 16 * COMP; i += 128)
        cfl[(i >> 8) * CF_STRIDE + (i & 255)] = src[i];
    __syncthreads();

    if (tid < CONV_OC * 12) {
        const int oc = tid / 12, p = tid % 12;
        float s = conv_b[oc];
        for (int ic = 0; ic < COMP; ++ic) {
            const float* wp = conv_w + (oc * COMP + ic) * CONV_K;
            const float* xp = cfl + ic;
            #pragma unroll
            for (int kk = 0; kk < CONV_K; ++kk)
                s += xp[(p + kk) * CF_STRIDE] * wp[kk];
        }
        convl[tid] = s;
    }
    __syncthreads();

    float r = c_b[tid];
    const float* w = c_w + tid * (CONV_OC * 12);
    #pragma unroll 8
    for (int j = 0; j < CONV_OC * 12; ++j) r += convl[j] * w[j];
    res[(long long)b * OUT_F + tid] = r;
}

// ---------------------------------------------------------------------------
extern "C" void kernel_launch(void* const* d_in, const int* in_sizes, int n_in,
                              void* d_out, int out_size, void* d_ws, size_t ws_size,
                              hipStream_t stream)
{
    const float* x      = (const float*)d_in[0];   // (4096, 4096)
    const float* W_caps = (const float*)d_in[1];   // (16, 4096, 512)
    const float* b_caps = (const float*)d_in[2];   // (16, 512)
    const float* QKV_w  = (const float*)d_in[3];   // (512, 768)
    const float* conv_w = (const float*)d_in[4];   // (8, 256, 5)
    const float* conv_b = (const float*)d_in[5];   // (8,)
    const float* c_w    = (const float*)d_in[6];   // (128, 96)
    const float* c_b    = (const float*)d_in[7];   // (128,)

    // d_out = [res (B*128) | latent_n (B*16*512) | cf (B*16*256)]
    float* res    = (float*)d_out;
    float* latent = res + (long long)B_SZ * OUT_F;
    float* cf     = latent + (long long)B_SZ * N_CAPS * CAP;

    float* qkv = (float*)d_ws;   // needs B*16*768*4 = 192 MB scratch

    // 1) latent_raw = relu(x @ W_caps[k] + b_caps[k]) -> interleaved (b,k,d)
    wmma_gemm_f32<<<dim3(B_SZ / TM, CAP / TN, N_CAPS), 256, 0, stream>>>(
        x, IN_F,
        W_caps, CAP, (long long)IN_F * CAP,
        latent, N_CAPS * CAP, (long long)CAP,
        b_caps, CAP,
        IN_F, /*relu=*/1);

    // 2) latent_n = l2norm(latent) in place (rows of 512)
    l2norm_rows<<<B_SZ * N_CAPS, 128, 0, stream>>>(latent, CAP);

    // 3) qkv = latent_n @ QKV_w  (65536 x 512 x 768)
    wmma_gemm_f32<<<dim3(B_SZ * N_CAPS / TM, QKV_E / TN, 1), 256, 0, stream>>>(
        latent, CAP,
        QKV_w, QKV_E, 0,
        qkv, QKV_E, 0,
        nullptr, 0,
        CAP, /*relu=*/0);

    // 4) attention + cf l2norm (one wave per batch, 4 per block)
    attn_kernel<<<B_SZ / 4, 128, 0, stream>>>(qkv, cf);

    // 5) conv1d + dense head
    conv_head<<<B_SZ, 128, 0, stream>>>(cf, conv_w, conv_b, c_w, c_b, res);
}